// SAGE_3040836846097
// MI455X (gfx1250) — compile-verified
//
#include <hip/hip_runtime.h>
#include <cstddef>

typedef __attribute__((ext_vector_type(2))) float v2f;
typedef __attribute__((ext_vector_type(8))) float v8f;

#define DDIM 128

// ---------------------------------------------------------------------------
// Stripe GEMM via V_WMMA_F32_16X16X4_F32 (fp32, matches reference precision).
// One wave computes a full 16-row x 128-col output stripe:
//   out[r,:] = act( A1[r,:] @ B1  (+ A2[r,:] @ B2)  + bias )
// A layout (16x4 f32): lane L holds A[r0 + (L&15)][k0 + 2*(L>>4) + {0,1}]
// B layout (4x16 f32): lane L holds B[k0 + 2*(L>>4) + {0,1}][t*16 + (L&15)]
// C/D layout: VGPR r, lanes 0-15 -> row r0+r, lanes 16-31 -> row r0+r+8.
// ---------------------------------------------------------------------------
template <bool DUAL, bool RELU>
__global__ void gemm_stripe_kernel(const float* __restrict__ A1,
                                   const float* __restrict__ B1,
                                   const float* __restrict__ A2,
                                   const float* __restrict__ B2,
                                   const float* __restrict__ bias,
                                   float* __restrict__ out,
                                   int nstripes) {
  const int stripe = blockIdx.x * (blockDim.x >> 5) + (threadIdx.x >> 5);
  if (stripe >= nstripes) return;          // whole-wave exit; EXEC stays all-1
  const int lane = threadIdx.x & 31;
  const int m    = lane & 15;
  const int hi   = lane >> 4;
  const int kb   = hi * 2;                 // 0 or 2
  const int r0   = stripe * 16;

  const float* a1row = A1 + (size_t)(r0 + m) * DDIM;
  const float* a2row = DUAL ? (A2 + (size_t)(r0 + m) * DDIM) : A1;

  v8f acc[8] = {};                         // 8 col tiles x 8 VGPRs

  for (int k0 = 0; k0 < DDIM; k0 += 4) {
    v2f a = *(const v2f*)(a1row + k0 + kb);        // aligned b64 load
#pragma unroll
    for (int t = 0; t < 8; ++t) {
      const int n = t * 16 + m;
      v2f b;
      b.x = B1[(k0 + kb) * DDIM + n];
      b.y = B1[(k0 + kb + 1) * DDIM + n];
      acc[t] = __builtin_amdgcn_wmma_f32_16x16x4_f32(
          false, a, false, b, (short)0, acc[t], false, false);
    }
    if (DUAL) {
      v2f a2 = *(const v2f*)(a2row + k0 + kb);
#pragma unroll
      for (int t = 0; t < 8; ++t) {
        const int n = t * 16 + m;
        v2f b;
        b.x = B2[(k0 + kb) * DDIM + n];
        b.y = B2[(k0 + kb + 1) * DDIM + n];
        acc[t] = __builtin_amdgcn_wmma_f32_16x16x4_f32(
            false, a2, false, b, (short)0, acc[t], false, false);
      }
    }
  }

#pragma unroll
  for (int t = 0; t < 8; ++t) {
    const int col = t * 16 + m;
    const float bv = bias[col];
#pragma unroll
    for (int r = 0; r < 8; ++r) {
      const int row = r0 + r + hi * 8;
      float v = acc[t][r] + bv;
      if (RELU) v = fmaxf(v, 0.0f);
      out[(size_t)row * DDIM + col] = v;
    }
  }
}

// ---------------------------------------------------------------------------
// CSR build (once per call, reused by all 3 layers).
// ---------------------------------------------------------------------------
__global__ void fill_zero_int_kernel(int* __restrict__ p, int n) {
  int i = blockIdx.x * blockDim.x + threadIdx.x;
  if (i < n) p[i] = 0;
}

__global__ void csr_count_kernel(const int* __restrict__ dst,
                                 int* __restrict__ cnt, int E) {
  int e = blockIdx.x * blockDim.x + threadIdx.x;
  if (e < E) atomicAdd(&cnt[dst[e]], 1);
}

// Single-block exclusive scan: per-thread serial chunk sums, one 1024-wide
// Hillis-Steele pass, then per-thread serial prefix writeback. 21 barriers.
__global__ void csr_scan_kernel(const int* __restrict__ cnt,
                                int* __restrict__ rowptr,
                                int* __restrict__ cur, int N) {
  __shared__ int ssum[1024];
  const int t = threadIdx.x;
  const int chunk = (N + 1023) >> 10;
  const int lo = min(t * chunk, N);
  const int hi = min(lo + chunk, N);
  int s = 0;
  for (int i = lo; i < hi; ++i) s += cnt[i];
  ssum[t] = s;
  __syncthreads();
  for (int off = 1; off < 1024; off <<= 1) {
    int v = (t >= off) ? ssum[t - off] : 0;
    __syncthreads();
    ssum[t] += v;
    __syncthreads();
  }
  int run = (t == 0) ? 0 : ssum[t - 1];
  for (int i = lo; i < hi; ++i) {
    rowptr[i] = run;
    cur[i] = run;
    run += cnt[i];
  }
  if (t == 1023) rowptr[N] = ssum[1023];
}

__global__ void csr_fill_kernel(const int* __restrict__ src,
                                const int* __restrict__ dst,
                                int* __restrict__ cur,
                                int* __restrict__ eidx, int E) {
  int e = blockIdx.x * blockDim.x + threadIdx.x;
  if (e < E) {
    int p = atomicAdd(&cur[dst[e]], 1);
    eidx[p] = src[e];
  }
}

// ---------------------------------------------------------------------------
// Segment-max as a gather: one wave per destination node, each lane owns a
// float4 (128 cols / 32 lanes). Register max-accumulate, single plain store.
// Zero init == DGL "no in-edge -> 0" rule (messages are >= 0 post-ReLU), and
// max is order-independent so CSR slot nondeterminism is harmless.
// ---------------------------------------------------------------------------
__global__ void segmax_gather_kernel(const float* __restrict__ pooled,
                                     const int* __restrict__ rowptr,
                                     const int* __restrict__ eidx,
                                     float* __restrict__ neigh, int N) {
  const int d = blockIdx.x * (blockDim.x >> 5) + (threadIdx.x >> 5);
  if (d >= N) return;
  const int lane = threadIdx.x & 31;
  const int beg = rowptr[d];
  const int end = rowptr[d + 1];
  float4 m = make_float4(0.f, 0.f, 0.f, 0.f);
  int e = beg;
  // 2-way unroll: keep two gathers in flight
  for (; e + 1 < end; e += 2) {
    const int s0 = eidx[e];
    const int s1 = eidx[e + 1];
    float4 v0 = ((const float4*)(pooled + (size_t)s0 * DDIM))[lane];
    float4 v1 = ((const float4*)(pooled + (size_t)s1 * DDIM))[lane];
    m.x = fmaxf(m.x, fmaxf(v0.x, v1.x));
    m.y = fmaxf(m.y, fmaxf(v0.y, v1.y));
    m.z = fmaxf(m.z, fmaxf(v0.z, v1.z));
    m.w = fmaxf(m.w, fmaxf(v0.w, v1.w));
  }
  if (e < end) {
    const int s0 = eidx[e];
    float4 v0 = ((const float4*)(pooled + (size_t)s0 * DDIM))[lane];
    m.x = fmaxf(m.x, v0.x);
    m.y = fmaxf(m.y, v0.y);
    m.z = fmaxf(m.z, v0.z);
    m.w = fmaxf(m.w, v0.w);
  }
  ((float4*)(neigh + (size_t)d * DDIM))[lane] = m;
}

// ---------------------------------------------------------------------------
// Per-graph mean: node2graph is sorted, so block g binary-searches its node
// range and reduces its 128 columns without atomics.
// ---------------------------------------------------------------------------
__device__ __forceinline__ int lower_bound_i(const int* __restrict__ a, int n,
                                             int key) {
  int lo = 0, hi = n;
  while (lo < hi) {
    int mid = (lo + hi) >> 1;
    if (a[mid] < key) lo = mid + 1; else hi = mid;
  }
  return lo;
}

__global__ void graph_mean_kernel(const float* __restrict__ H,
                                  const int* __restrict__ n2g,
                                  float* __restrict__ gmean, int N) {
  const int g = blockIdx.x;
  const int lo = lower_bound_i(n2g, N, g);
  const int hi = lower_bound_i(n2g, N, g + 1);
  const int t = threadIdx.x;               // 0..127 = column
  float s = 0.0f;
  for (int i = lo; i < hi; ++i) s += H[(size_t)i * DDIM + t];
  const float c = (float)(hi - lo);
  gmean[(size_t)g * DDIM + t] = s / fmaxf(c, 1.0f);
}

// ---------------------------------------------------------------------------
extern "C" void kernel_launch(void* const* d_in, const int* in_sizes, int n_in,
                              void* d_out, int out_size, void* d_ws,
                              size_t ws_size, hipStream_t stream) {
  const float* h      = (const float*)d_in[0];
  const int*   src    = (const int*)d_in[1];
  const int*   dst    = (const int*)d_in[2];
  const int*   n2g    = (const int*)d_in[3];
  const float* W_pool = (const float*)d_in[4];
  const float* b_pool = (const float*)d_in[5];
  const float* W_self = (const float*)d_in[6];
  const float* W_ngh  = (const float*)d_in[7];
  const float* b_sage = (const float*)d_in[8];
  const float* W_mlp  = (const float*)d_in[9];
  const float* b_mlp  = (const float*)d_in[10];

  const int N = in_sizes[0] / DDIM;                 // 50000
  const int E = in_sizes[1];                        // 800000
  const int L = in_sizes[4] / (DDIM * DDIM);        // 3
  const int G = out_size / DDIM - N;                // 64

  float* Hout   = (float*)d_out;                    // N*D local_feat (in-place)
  float* gout   = Hout + (size_t)N * DDIM;          // G*D global_feat
  float* pooled = (float*)d_ws;                     // N*D
  float* neigh  = pooled + (size_t)N * DDIM;        // N*D
  float* gmean  = neigh + (size_t)N * DDIM;         // G*D
  int*   cnt    = (int*)(gmean + (size_t)G * DDIM); // N
  int*   rowptr = cnt + N;                          // N+1
  int*   cur    = rowptr + (N + 1);                 // N
  int*   eidx   = cur + N;                          // E

  const int nstripes = N / 16;                      // 3125 (N % 16 == 0)
  const dim3 gemm_grid((nstripes + 7) / 8);
  const dim3 edge_grid((E + 255) / 256);
  const dim3 node_wave_grid((N + 7) / 8);

  // ---- Build CSR by destination (once; reused by all layers) ----
  fill_zero_int_kernel<<<(N + 255) / 256, 256, 0, stream>>>(cnt, N);
  csr_count_kernel<<<edge_grid, 256, 0, stream>>>(dst, cnt, E);
  csr_scan_kernel<<<1, 1024, 0, stream>>>(cnt, rowptr, cur, N);
  csr_fill_kernel<<<edge_grid, 256, 0, stream>>>(src, dst, cur, eidx, E);

  // ---- 3 SAGE('pool') layers ----
  const float* Hin = h;
  for (int l = 0; l < L; ++l) {
    // pooled = relu(Hin @ W_pool[l] + b_pool[l])
    gemm_stripe_kernel<false, true><<<gemm_grid, 256, 0, stream>>>(
        Hin, W_pool + (size_t)l * DDIM * DDIM, nullptr, nullptr,
        b_pool + (size_t)l * DDIM, pooled, nstripes);
    // neigh[d] = max over in-edges of pooled[src] (atomic-free gather)
    segmax_gather_kernel<<<node_wave_grid, 256, 0, stream>>>(
        pooled, rowptr, eidx, neigh, N);
    // Hout = relu(Hin @ W_self[l] + neigh @ W_neigh[l] + b_sage[l])
    gemm_stripe_kernel<true, true><<<gemm_grid, 256, 0, stream>>>(
        Hin, W_self + (size_t)l * DDIM * DDIM, neigh,
        W_ngh + (size_t)l * DDIM * DDIM, b_sage + (size_t)l * DDIM, Hout,
        nstripes);
    Hin = Hout;                                     // in-place thereafter
  }

  // ---- readout ----
  graph_mean_kernel<<<G, DDIM, 0, stream>>>(Hout, n2g, gmean, N);
  gemm_stripe_kernel<false, false><<<1, 256, 0, stream>>>(
      gmean, W_mlp, nullptr, nullptr, b_mlp, gout, G / 16);
}